// BRNNIntegrateOnehot_76751065579810
// MI455X (gfx1250) — compile-verified
//
#include <hip/hip_runtime.h>
#include <stdint.h>

// ---------------------------------------------------------------------------
// BRNN FSA integrate: h_{t+1} = clip(h_t @ fsa[tok[b,t]]), out[b,t,:] = h_{t+1}
// B=64, L=512, S=128, fsa table = 10000 x 128 x 128 f32 (655 MB).
// Memory-bound random-gather: one 64KB matrix tile per (b,t). Strategy:
//   * 1 workgroup per batch element, 8 waves (wave32) = 256 threads
//   * TDM tensor_load_to_lds streams the next step's 64KB tile into LDS
//     (double-buffered) while WMMA consumes the current one
//   * V_WMMA_F32_16X16X4_F32 does the vec-mat (h replicated over A rows)
// ---------------------------------------------------------------------------

typedef __attribute__((ext_vector_type(2))) float        v2f;
typedef __attribute__((ext_vector_type(8))) float        v8f;
typedef __attribute__((ext_vector_type(4))) unsigned int v4u;
typedef __attribute__((ext_vector_type(4))) int          v4i;
typedef __attribute__((ext_vector_type(8))) int          v8i;

#define S_DIM       128
#define ROW_PAD_DW  8                        // TDM pads 8 dwords per 128-dword row
#define ROW_STRIDE  (S_DIM + ROW_PAD_DW)     // 136 dwords -> rows k, k+2 hit disjoint banks
#define MAT_DWORDS  (S_DIM * ROW_STRIDE)     // 17408
#define MAT_BYTES   (MAT_DWORDS * 4)         // 69632 bytes per LDS tile buffer

// Issue one TDM DMA: 128x128 f32 tile (row-major, tensor stride 128) from
// global `gaddr` into LDS byte offset `lds_off`, padded to 136-dword rows.
__device__ __forceinline__ void tdm_load_tile(unsigned lds_off, const float* gaddr) {
    uint64_t ga = (uint64_t)(uintptr_t)gaddr;
    v4u g0;
    g0[0] = 1u;                                   // count=1, user descriptor, no gather
    g0[1] = lds_off;                              // lds_addr (bytes)
    g0[2] = (unsigned)ga;                         // global_addr[31:0]
    g0[3] = ((unsigned)(ga >> 32) & 0x01FFFFFFu)  // global_addr[56:32]
          | 0x80000000u;                          // type=2 ("image") at bits 127:126
    v8i g1;
    g1[0] = (2 << 16)                             // data_size = 4 bytes
          | (1 << 20)                             // pad_enable
          | (6 << 22)                             // pad_interval: every 128 dwords
          | (7 << 25);                            // pad_amount: 8 dwords
    g1[1] = (S_DIM << 16);                        // tensor_dim0 lo16 = 128
    g1[2] = (S_DIM << 16);                        // tensor_dim0 hi=0 | tensor_dim1 lo16 = 128
    g1[3] = (S_DIM << 16);                        // tensor_dim1 hi=0 | tile_dim0 = 128
    g1[4] = S_DIM;                                // tile_dim1 = 128, tile_dim2 = 0 (2D)
    g1[5] = S_DIM;                                // tensor_dim0_stride = 128 elements
    g1[6] = 0;
    g1[7] = 0;
    v4i z4 = {0, 0, 0, 0};
#if defined(__clang_major__) && (__clang_major__ >= 23)
    v8i z8 = {0, 0, 0, 0, 0, 0, 0, 0};
    __builtin_amdgcn_tensor_load_to_lds(g0, g1, z4, z4, z8, 0);
#else
    __builtin_amdgcn_tensor_load_to_lds(g0, g1, z4, z4, 0);
#endif
}

__global__ __launch_bounds__(256, 1)
void brnn_fsa_kernel(const int* __restrict__ tokens,   // [B, L]
                     const float* __restrict__ fsa,    // [V, 128, 128]
                     float* __restrict__ out,          // [B, L, 128]
                     int L) {
    extern __shared__ float smem[];
    float* mat0 = smem;                 // padded tile, 17408 dwords
    float* mat1 = smem + MAT_DWORDS;
    float* hbuf = smem + 2 * MAT_DWORDS; // h double buffer: 2 x 128

    const int b    = blockIdx.x;
    const int tid  = threadIdx.x;
    const int lane = tid & 31;
    const int wave = tid >> 5;          // 0..7 == N-tile index
    const unsigned lds_base = (unsigned)(uintptr_t)smem; // LDS byte offset = low 32 bits

    // h0 = one-hot(state 0)
    if (tid < S_DIM) {
        hbuf[tid]         = (tid == 0) ? 1.0f : 0.0f;
        hbuf[S_DIM + tid] = 0.0f;
    }

    // Prologue: DMA tile for t=0 into buffer 0 (wave 0 issues; TDM ignores EXEC)
    if (wave == 0) {
        int tok0 = tokens[(size_t)b * L];                     // scalar (uniform) load
        tdm_load_tile(lds_base, fsa + (size_t)tok0 * (S_DIM * S_DIM));
        __builtin_amdgcn_s_wait_tensorcnt(0);
    }
    __syncthreads();

    // WMMA f32 16x16x4 operand layout (wave32):
    //   A vgpr0: lanes0-15 K=0 | lanes16-31 K=2 ; vgpr1: K=1 | K=3
    //   B vgpr0: lanes0-15 row K=0, N=lane | lanes16-31 row K=2 ; vgpr1: K=1 | K=3
    const int ksel = (lane >= 16) ? 2 : 0;
    const int col  = wave * 16 + (lane & 15);

    for (int t = 0; t < L; ++t) {
        const float* mat  = (t & 1) ? mat1 : mat0;
        const float* hin  = hbuf + (t & 1) * S_DIM;
        float*       hout = hbuf + ((t + 1) & 1) * S_DIM;

        // Prefetch next tile into the other buffer while we compute this step.
        if (wave == 0 && (t + 1) < L) {
            int tok = tokens[(size_t)b * L + t + 1];
            tdm_load_tile(lds_base + ((unsigned)((t + 1) & 1)) * MAT_BYTES,
                          fsa + (size_t)tok * (S_DIM * S_DIM));
        }

        // C[m][n] = sum_k h[k]*Tr[k][n]  (h replicated over the 16 A-rows,
        // so every C row equals the result). Two accumulators halve the
        // serial WMMA D->C dependency chain.
        v8f c0 = {}; v8f c1 = {};
#pragma unroll
        for (int kb = 0; kb < 32; kb += 2) {
            int k0 = kb * 4 + ksel;
            v2f a0; a0.x = hin[k0];     a0.y = hin[k0 + 1];
            v2f b0; b0.x = mat[k0 * ROW_STRIDE + col];
                    b0.y = mat[(k0 + 1) * ROW_STRIDE + col];
            c0 = __builtin_amdgcn_wmma_f32_16x16x4_f32(false, a0, false, b0,
                                                       (short)0, c0, false, false);
            int k1 = (kb + 1) * 4 + ksel;
            v2f a1; a1.x = hin[k1];     a1.y = hin[k1 + 1];
            v2f b1; b1.x = mat[k1 * ROW_STRIDE + col];
                    b1.y = mat[(k1 + 1) * ROW_STRIDE + col];
            c1 = __builtin_amdgcn_wmma_f32_16x16x4_f32(false, a1, false, b1,
                                                       (short)0, c1, false, false);
        }

        // D vgpr0, lanes 0-15: M=0, N=lane -> result element (wave*16 + lane)
        float r = c0[0] + c1[0];
        r = fminf(fmaxf(r, -10.0f), 10.0f);
        if (lane < 16) {
            out[((size_t)b * L + t) * S_DIM + col] = r;
            hout[col] = r;
        }

        // Ensure next tile landed before anyone reads it, then rendezvous.
        if (wave == 0) __builtin_amdgcn_s_wait_tensorcnt(0);
        __syncthreads();
    }
}

extern "C" void kernel_launch(void* const* d_in, const int* in_sizes, int n_in,
                              void* d_out, int out_size, void* d_ws, size_t ws_size,
                              hipStream_t stream) {
    const int*   tokens = (const int*)d_in[0];   // input [B, L]
    // d_in[1] = lengths (unused by the reference computation)
    const float* fsa    = (const float*)d_in[2]; // fsa_tensor [V, 128, 128]
    float*       out    = (float*)d_out;         // [B, L, 128] f32

    const int B = in_sizes[1];
    const int L = in_sizes[0] / B;

    const size_t shmem = 2 * MAT_BYTES + 2 * S_DIM * sizeof(float); // ~137 KB of 320 KB WGP LDS
    brnn_fsa_kernel<<<dim3(B), dim3(256), shmem, stream>>>(tokens, fsa, out, L);
}